// LSTMDecoder_27255862461070
// MI455X (gfx1250) — compile-verified
//
#include <hip/hip_runtime.h>
#include <math.h>

// Problem constants (from reference)
#define VOCAB 32000
#define EMB   512
#define DEC_H 1024
#define ATTN  512
#define BB    64
#define SS    128
#define ENC2  1024

#define CK      128   // K-chunk staged in LDS for M=64 GEMMs
#define LDA_PAD 132   // LDS row stride (floats); 132 % 64 == 4 -> conflict-free b64 reads

typedef __attribute__((ext_vector_type(2))) float v2f;
typedef __attribute__((ext_vector_type(8))) float v8f;

#define WMMA_F32(a, b, c) \
  __builtin_amdgcn_wmma_f32_16x16x4_f32(false, (a), false, (b), (short)0, (c), false, false)

// ---------------------------------------------------------------------------
// Generic tiling (used for enc_proj, M=8192): wave computes 16Mx64N.
// A [M,K] row-major, W [N,K] row-major, C = A @ W^T.
// Fragment layouts per CDNA5 ISA 7.12.2 (32-bit A 16x4 / B 4x16):
//   lanes 0-15 carry K=k0..k0+1 (frag .x/.y), lanes 16-31 carry K=k0+2..k0+3.
// ---------------------------------------------------------------------------
__global__ void wmma_gemm_generic_kernel(const float* __restrict__ A, const float* __restrict__ W,
                                         int K, float* __restrict__ C, int M, int N, int ldC) {
  const int wave = (blockIdx.x * blockDim.x + threadIdx.x) >> 5;
  const int lane = threadIdx.x & 31;
  const int nT = N >> 6;
  const int mTile = wave / nT;
  const int nTile = wave - mTile * nT;
  if (mTile >= (M >> 4)) return;               // whole-wave uniform exit
  const int m0 = mTile << 4, n0 = nTile << 6;
  const int l15  = lane & 15;
  const int koff = (lane >> 4) << 1;

  v8f acc[4] = {};
  const float* Ar = A + (size_t)(m0 + l15) * K + koff;
  const float* W0 = W + (size_t)(n0 + l15) * K + koff;
  const float* W1 = W0 + (size_t)16 * K;
  const float* W2 = W0 + (size_t)32 * K;
  const float* W3 = W0 + (size_t)48 * K;
#pragma unroll 2
  for (int k = 0; k < K; k += 4) {
    v2f a  = *(const v2f*)(Ar + k);
    v2f b0 = *(const v2f*)(W0 + k);
    v2f b1 = *(const v2f*)(W1 + k);
    v2f b2 = *(const v2f*)(W2 + k);
    v2f b3 = *(const v2f*)(W3 + k);
    acc[0] = WMMA_F32(a, b0, acc[0]);
    acc[1] = WMMA_F32(a, b1, acc[1]);
    acc[2] = WMMA_F32(a, b2, acc[2]);
    acc[3] = WMMA_F32(a, b3, acc[3]);
  }

  const int mhalf = (lane >> 4) << 3;
#pragma unroll
  for (int t = 0; t < 4; ++t) {
    const int n = n0 + t * 16 + l15;
#pragma unroll
    for (int r = 0; r < 8; ++r)
      C[(size_t)(m0 + mhalf + r) * ldC + n] = acc[t][r];
  }
}

// Issue one 64xCK A-chunk into LDS buffer: 2048 x 16B transfers, 8 per thread,
// via async global->LDS DMA (tracked by ASYNCcnt; 8 async instrs per wave).
__device__ __forceinline__ void stage_chunk_async(const float* __restrict__ A, int K, int kb,
                                                  unsigned ldsBuf, int tid) {
#pragma unroll
  for (int i = 0; i < 8; ++i) {
    int id = tid + i * 256;                 // float4 slot 0..2047
    int r  = id >> 5;                       // row 0..63
    int kk = (id & 31) << 2;                // col 0..124
    unsigned lds = ldsBuf + (unsigned)(r * LDA_PAD + kk) * 4u;
    const float* g = A + (size_t)r * K + kb + kk;
    asm volatile("global_load_async_to_lds_b128 %0, %1, off"
                 :: "v"(lds), "v"((unsigned long long)(uintptr_t)g)
                 : "memory");
  }
}

// ---------------------------------------------------------------------------
// M=64 GEMM: each wave covers ALL 64 M-rows x 32 N-cols (8 accumulators), so
// every weight element streams from HBM exactly once (critical for the
// 327 MB W_out). A is double-buffered through LDS in 64xCK chunks via
// global_load_async_to_lds_b128: chunk c+1 is issued before computing chunk c,
// and `s_wait_asynccnt 8` (async loads complete in order; 8 per wave per
// chunk) guarantees chunk c is resident while c+1 is still in flight.
// Supports a fused second pair (A2,W2) accumulating into the same tile.
// ---------------------------------------------------------------------------
__global__ void __launch_bounds__(256)
wmma_gemm_m64_kernel(const float* __restrict__ A1, const float* __restrict__ W1, int K1,
                     const float* __restrict__ A2, const float* __restrict__ W2, int K2,
                     const float* __restrict__ bias1, const float* __restrict__ bias2,
                     float* __restrict__ C, int N, int ldC) {
  __shared__ float tileA[2][64 * LDA_PAD];   // 66 KB of the 320 KB WGP LDS
  const int tid  = threadIdx.x;
  const int wave = tid >> 5;
  const int lane = tid & 31;
  const int l15  = lane & 15;
  const int koff = (lane >> 4) << 1;
  const int n0   = (blockIdx.x * 8 + wave) * 32;

  v8f acc[8] = {};
  const unsigned ldsB0 = (unsigned)(uintptr_t)&tileA[0][0];  // low 32b = LDS offset
  const unsigned ldsB1 = (unsigned)(uintptr_t)&tileA[1][0];

  for (int pass = 0; pass < 2; ++pass) {
    const float* A = pass ? A2 : A1;
    const float* W = pass ? W2 : W1;
    const int    K = pass ? K2 : K1;
    if (!A) continue;                                   // uniform branch
    const float* Wp0 = W + (size_t)(n0 + l15) * K + koff;
    const float* Wp1 = Wp0 + (size_t)16 * K;
    const int nch = K / CK;

    stage_chunk_async(A, K, 0, ldsB0, tid);             // prologue: chunk 0

    for (int c = 0; c < nch; ++c) {
      const unsigned bufCur = (c & 1) ? ldsB1 : ldsB0;
      const int kb = c * CK;
      if (c + 1 < nch) {
        stage_chunk_async(A, K, kb + CK, (c & 1) ? ldsB0 : ldsB1, tid);
        asm volatile("s_wait_asynccnt 0x8" ::: "memory");  // chunk c done, c+1 in flight
      } else {
        asm volatile("s_wait_asynccnt 0x0" ::: "memory");  // drain
      }
      __syncthreads();                                     // chunk c visible to all waves

      const float* As0 = (const float*)(uintptr_t)0 ;      // (unused; keep indices simple)
      (void)As0;
#pragma unroll 4
      for (int k = 0; k < CK; k += 4) {
        const float* As = &tileA[c & 1][0] + k + koff;
        v2f a0 = *(const v2f*)(As + (l15     ) * LDA_PAD);
        v2f a1 = *(const v2f*)(As + (l15 + 16) * LDA_PAD);
        v2f a2 = *(const v2f*)(As + (l15 + 32) * LDA_PAD);
        v2f a3 = *(const v2f*)(As + (l15 + 48) * LDA_PAD);
        v2f b0 = *(const v2f*)(Wp0 + kb + k);
        v2f b1 = *(const v2f*)(Wp1 + kb + k);
        acc[0] = WMMA_F32(a0, b0, acc[0]);  acc[1] = WMMA_F32(a0, b1, acc[1]);
        acc[2] = WMMA_F32(a1, b0, acc[2]);  acc[3] = WMMA_F32(a1, b1, acc[3]);
        acc[4] = WMMA_F32(a2, b0, acc[4]);  acc[5] = WMMA_F32(a2, b1, acc[5]);
        acc[6] = WMMA_F32(a3, b0, acc[6]);  acc[7] = WMMA_F32(a3, b1, acc[7]);
      }
      __syncthreads();   // all waves done with buffer c before it is re-staged (at c+2)
      (void)bufCur;
    }
  }

  // Epilogue: C/D layout lanes 0-15 -> M=r, lanes 16-31 -> M=r+8
  const int mhalf = (lane >> 4) << 3;
#pragma unroll
  for (int nt = 0; nt < 2; ++nt) {
    const int n = n0 + nt * 16 + l15;
    float bv = 0.f;
    if (bias1) bv += bias1[n];
    if (bias2) bv += bias2[n];
#pragma unroll
    for (int mt = 0; mt < 4; ++mt) {
      v8f a = acc[mt * 2 + nt];
#pragma unroll
      for (int r = 0; r < 8; ++r)
        C[(size_t)(mt * 16 + mhalf + r) * ldC + n] = a[r] + bv;
    }
  }
}

// Embedding lookup with padding_idx=0; writes into x[:, :EMB] and out_in[:, 2048:2560]
__global__ void embed_kernel(const int* __restrict__ tok, const float* __restrict__ emb,
                             float* __restrict__ x, float* __restrict__ out_in) {
  int gid = blockIdx.x * blockDim.x + threadIdx.x;
  if (gid >= BB * EMB) return;
  int b = gid >> 9, e = gid & 511;
  int t = tok[b];
  float v = (t == 0) ? 0.f : emb[(size_t)t * EMB + e];
  x[b * (EMB + ENC2) + e]     = v;
  out_in[b * 2560 + 2048 + e] = v;
}

// scores[b,s] = v . tanh(enc_proj[s,b,:] + dec_proj[b,:]); one wave per (s,b) row
__global__ void attn_score_kernel(const float* __restrict__ ep, const float* __restrict__ dp,
                                  const float* __restrict__ v, const int* __restrict__ mask,
                                  float* __restrict__ scores) {
  int wave = (blockIdx.x * blockDim.x + threadIdx.x) >> 5;
  int lane = threadIdx.x & 31;
  if (wave >= SS * BB) return;
  int s = wave >> 6, b = wave & 63;
  const float* eprow = ep + (size_t)wave * ATTN;
  const float* dprow = dp + (size_t)b * ATTN;
  float part = 0.f;
#pragma unroll
  for (int a = lane; a < ATTN; a += 32)
    part += v[a] * tanhf(eprow[a] + dprow[a]);
  for (int off = 16; off; off >>= 1) part += __shfl_xor(part, off, 32);
  if (lane == 0)
    scores[b * SS + s] = (mask[b * SS + s] == 0) ? -__builtin_inff() : part;
}

// Row softmax over S=128; one wave per batch row; writes attn_w into d_out slice
__global__ void softmax_kernel(const float* __restrict__ scores, float* __restrict__ attn) {
  int wave = (blockIdx.x * blockDim.x + threadIdx.x) >> 5;
  int lane = threadIdx.x & 31;
  if (wave >= BB) return;
  const float* row = scores + (size_t)wave * SS;
  float v0 = row[lane], v1 = row[lane + 32], v2 = row[lane + 64], v3 = row[lane + 96];
  float mx = fmaxf(fmaxf(v0, v1), fmaxf(v2, v3));
  for (int off = 16; off; off >>= 1) mx = fmaxf(mx, __shfl_xor(mx, off, 32));
  float e0 = expf(v0 - mx), e1 = expf(v1 - mx), e2 = expf(v2 - mx), e3 = expf(v3 - mx);
  float sum = e0 + e1 + e2 + e3;
  for (int off = 16; off; off >>= 1) sum += __shfl_xor(sum, off, 32);
  float inv = 1.f / sum;
  float* o = attn + (size_t)wave * SS;
  o[lane] = e0 * inv; o[lane + 32] = e1 * inv; o[lane + 64] = e2 * inv; o[lane + 96] = e3 * inv;
}

// context[b,e] = sum_s attn[b,s] * enc[s,b,e]; coalesced over e
__global__ void context_kernel(const float* __restrict__ attn, const float* __restrict__ enc,
                               float* __restrict__ x, float* __restrict__ out_in) {
  int gid = blockIdx.x * blockDim.x + threadIdx.x;
  if (gid >= BB * ENC2) return;
  int b = gid >> 10, e = gid & 1023;
  const float* arow = attn + (size_t)b * SS;
  float sum = 0.f;
#pragma unroll 4
  for (int s = 0; s < SS; ++s)
    sum += arow[s] * enc[((size_t)s * BB + b) * ENC2 + e];
  x[b * (EMB + ENC2) + EMB + e] = sum;
  out_in[b * 2560 + DEC_H + e]  = sum;
}

// LSTM cell pointwise (torch gate order i,f,g,o)
__global__ void lstm_kernel(const float* __restrict__ gates, const float* __restrict__ cell,
                            float* __restrict__ out_hid, float* __restrict__ out_cell,
                            float* __restrict__ out_in) {
  int gid = blockIdx.x * blockDim.x + threadIdx.x;
  if (gid >= BB * DEC_H) return;
  int b = gid >> 10, j = gid & 1023;
  const float* g = gates + (size_t)b * (4 * DEC_H);
  float i = 1.f / (1.f + expf(-g[j]));
  float f = 1.f / (1.f + expf(-g[j + DEC_H]));
  float c = tanhf(g[j + 2 * DEC_H]);
  float o = 1.f / (1.f + expf(-g[j + 3 * DEC_H]));
  float nc = f * cell[gid] + i * c;
  float nh = o * tanhf(nc);
  out_cell[gid] = nc;
  out_hid[gid]  = nh;
  out_in[b * 2560 + j] = nh;
}

extern "C" void kernel_launch(void* const* d_in, const int* in_sizes, int n_in,
                              void* d_out, int out_size, void* d_ws, size_t ws_size,
                              hipStream_t stream) {
  const int*   tok    = (const int*)d_in[0];
  const float* hidden = (const float*)d_in[1];
  const float* cell   = (const float*)d_in[2];
  const float* enc    = (const float*)d_in[3];
  const int*   mask   = (const int*)d_in[4];
  const float* emb    = (const float*)d_in[5];
  const float* W_enc  = (const float*)d_in[6];
  const float* W_dec  = (const float*)d_in[7];
  const float* v_attn = (const float*)d_in[8];
  const float* W_ih   = (const float*)d_in[9];
  const float* W_hh   = (const float*)d_in[10];
  const float* b_ih   = (const float*)d_in[11];
  const float* b_hh   = (const float*)d_in[12];
  const float* W_out  = (const float*)d_in[13];
  const float* b_out  = (const float*)d_in[14];

  float* out       = (float*)d_out;
  float* out_logit = out;                                   // [64,32000]
  float* out_hid   = out + (size_t)BB * VOCAB;              // [64,1024]
  float* out_cell  = out_hid + BB * DEC_H;                  // [64,1024]
  float* out_attn  = out_cell + BB * DEC_H;                 // [64,128]

  float* ws     = (float*)d_ws;
  float* x      = ws;                         // [64,1536]  (embedded|context)
  float* out_in = x + BB * (EMB + ENC2);      // [64,2560]  (hid|context|embedded)
  float* decp   = out_in + BB * 2560;         // [64,512]
  float* scores = decp + BB * ATTN;           // [64,128]
  float* gates  = scores + BB * SS;           // [64,4096]
  float* encp   = gates + BB * 4 * DEC_H;     // [8192,512]

  // 1. embedding lookup
  embed_kernel<<<(BB * EMB) / 256, 256, 0, stream>>>(tok, emb, x, out_in);

  // 2. dec_proj = hidden @ W_dec^T   [64,512]
  wmma_gemm_m64_kernel<<<ATTN / 256, 256, 0, stream>>>(
      hidden, W_dec, DEC_H, nullptr, nullptr, 0, nullptr, nullptr,
      decp, ATTN, ATTN);

  // 3. enc_proj = enc_flat @ W_enc^T  [8192,512]  (A streamed once, W cached)
  wmma_gemm_generic_kernel<<<((SS * BB) / 16) * (ATTN / 64) * 32 / 256, 256, 0, stream>>>(
      enc, W_enc, ENC2, encp, SS * BB, ATTN, ATTN);

  // 4. scores = v . tanh(enc_proj + dec_proj), masked
  attn_score_kernel<<<(SS * BB) * 32 / 256, 256, 0, stream>>>(encp, decp, v_attn, mask, scores);

  // 5. softmax -> attn_w (written directly into d_out slice)
  softmax_kernel<<<(BB * 32 + 255) / 256, 256, 0, stream>>>(scores, out_attn);

  // 6. context = attn_w @ enc (per-batch)
  context_kernel<<<(BB * ENC2) / 256, 256, 0, stream>>>(out_attn, enc, x, out_in);

  // 7. gates = x @ W_ih^T + hidden @ W_hh^T + b_ih + b_hh   [64,4096]
  wmma_gemm_m64_kernel<<<(4 * DEC_H) / 256, 256, 0, stream>>>(
      x, W_ih, EMB + ENC2, hidden, W_hh, DEC_H, b_ih, b_hh,
      gates, 4 * DEC_H, 4 * DEC_H);

  // 8. LSTM pointwise -> new_hidden / new_cell (d_out) and out_in[:, :1024]
  lstm_kernel<<<(BB * DEC_H) / 256, 256, 0, stream>>>(gates, cell, out_hid, out_cell, out_in);

  // 9. logits = out_in @ W_out^T + b_out  [64,32000]; W_out streams HBM exactly once
  wmma_gemm_m64_kernel<<<VOCAB / 256, 256, 0, stream>>>(
      out_in, W_out, DEC_H + ENC2 + EMB, nullptr, nullptr, 0, b_out, nullptr,
      out_logit, VOCAB, VOCAB);
}